// NCP_FC_11570641895536
// MI455X (gfx1250) — compile-verified
//
#include <hip/hip_runtime.h>
#include <hip/hip_bf16.h>

#define B_      256
#define T_      2048
#define S_      32
#define N_      19
#define MOTOR_  1
#define CLASSES_ 10
#define UNFOLDS_ 6
#define EPS_    1e-8f

typedef __attribute__((ext_vector_type(2))) float v2f;
typedef __attribute__((ext_vector_type(8))) float v8f;

__device__ __forceinline__ float fast_sigmoid(float x) {
    // 1/(1+exp(-x)) = rcp(1 + exp2(-x*log2(e)))
    float e = __builtin_amdgcn_exp2f(-x * 1.44269504088896341f);
    return __builtin_amdgcn_rcpf(1.0f + e);
}

__device__ __forceinline__ float lane_bcast(float v, int lane) {
    return __uint_as_float(__builtin_amdgcn_readlane(__float_as_uint(v), lane));
}

// ---------------------------------------------------------------------------
// Kernel 1 (parallel): precompute sensory synapse sums w_num_s / w_den_s
// for every (b, t, n).  No recurrence -> full GPU parallelism.
// ---------------------------------------------------------------------------
__global__ void sensory_precompute(const float* __restrict__ x,
                                   const float* __restrict__ input_w,
                                   const float* __restrict__ input_b,
                                   const float* __restrict__ smu,
                                   const float* __restrict__ ssig,
                                   const float* __restrict__ sw,
                                   const float* __restrict__ serev,
                                   const int*   __restrict__ smask,
                                   float* __restrict__ wns,
                                   float* __restrict__ wds) {
    size_t tid = (size_t)blockIdx.x * blockDim.x + threadIdx.x;
    const size_t total = (size_t)B_ * T_ * N_;
    if (tid >= total) return;
    int    n  = (int)(tid % N_);
    size_t bt = tid / N_;
    const float* xr = x + bt * S_;
    float num = 0.0f, den = 0.0f;
#pragma unroll
    for (int s = 0; s < S_; ++s) {
        float inp = fmaf(xr[s], input_w[s], input_b[s]);
        float sg  = fast_sigmoid(ssig[s * N_ + n] * (inp - smu[s * N_ + n]));
        float wm  = sw[s * N_ + n] * (float)smask[s * N_ + n];
        num = fmaf(wm * serev[s * N_ + n], sg, num);
        den = fmaf(wm, sg, den);
    }
    wns[tid] = num;
    wds[tid] = den;
}

// ---------------------------------------------------------------------------
// Kernel 2 (serial scan): one wave32 per batch element.
// Lanes = post-synaptic neurons (19 active); presynaptic state broadcast
// with v_readlane.  All (pre,post) parameters held in unrolled register
// arrays.  PRE=true reads precomputed sensory sums; PRE=false computes them
// inline (fallback when workspace is small).
// ---------------------------------------------------------------------------
template <bool PRE>
__global__ __launch_bounds__(32)
void ltc_scan(const float* __restrict__ x,
              const float* __restrict__ input_w,
              const float* __restrict__ input_b,
              const float* __restrict__ smu,
              const float* __restrict__ ssig,
              const float* __restrict__ sw,
              const float* __restrict__ serev,
              const int*   __restrict__ smask,
              const float* __restrict__ mu,
              const float* __restrict__ sigma,
              const float* __restrict__ w,
              const float* __restrict__ erev,
              const int*   __restrict__ mask,
              const float* __restrict__ gleak,
              const float* __restrict__ vleak,
              const float* __restrict__ cm,
              const float* __restrict__ output_w,
              const float* __restrict__ output_b,
              const float* __restrict__ wns,
              const float* __restrict__ wds,
              float* __restrict__ outs) {
    const int b    = blockIdx.x;
    const int lane = threadIdx.x;
    const int n    = (lane < N_) ? lane : (N_ - 1);   // clamp idle lanes

    // Per-edge recurrent parameters: lane = post neuron, array idx = pre.
    float sig_r[N_], sm_r[N_], w_r[N_], we_r[N_];
#pragma unroll
    for (int p = 0; p < N_; ++p) {
        float sg = sigma[p * N_ + n];
        sig_r[p] = sg;
        sm_r[p]  = sg * mu[p * N_ + n];
        float wm = w[p * N_ + n] * (float)mask[p * N_ + n];
        w_r[p]   = wm;
        we_r[p]  = wm * erev[p * N_ + n];
    }
    const float gl   = gleak[n];
    const float glvl = gl * vleak[n];
    const float cmt  = cm[n] * (float)UNFOLDS_;
    const float ow   = output_w[0];
    const float ob   = output_b[0];

    // Fused-sensory parameters (only when !PRE): arrays over s.
    float ssig_r[S_], ssm_r[S_], sw_r[S_], swe_r[S_];
    float iw = 0.0f, ib = 0.0f;
    if (!PRE) {
#pragma unroll
        for (int s = 0; s < S_; ++s) {
            float a  = ssig[s * N_ + n];
            ssig_r[s] = a;
            ssm_r[s]  = a * smu[s * N_ + n];
            float wm  = sw[s * N_ + n] * (float)smask[s * N_ + n];
            sw_r[s]   = wm;
            swe_r[s]  = wm * serev[s * N_ + n];
        }
        iw = input_w[lane];    // lane in [0,32) == S_
        ib = input_b[lane];
    }

    float v = 0.0f;
    for (int t = 0; t < T_; ++t) {
        float wn_s, wd_s;
        if (PRE) {
            size_t idx = ((size_t)b * T_ + t) * N_ + n;
            wn_s = wns[idx];
            wd_s = wds[idx];
        } else {
            float xr = x[((size_t)b * T_ + t) * S_ + lane];
            float xa = fmaf(xr, iw, ib);     // per-sensory affine, lane = s
            wn_s = 0.0f; wd_s = 0.0f;
#pragma unroll
            for (int s = 0; s < S_; ++s) {
                float xs = lane_bcast(xa, s);
                float sg = fast_sigmoid(fmaf(ssig_r[s], xs, -ssm_r[s]));
                wn_s = fmaf(swe_r[s], sg, wn_s);
                wd_s = fmaf(sw_r[s], sg, wd_s);
            }
        }
#pragma unroll
        for (int u = 0; u < UNFOLDS_; ++u) {
            float an = wn_s, ad = wd_s;
#pragma unroll
            for (int p = 0; p < N_; ++p) {
                float vp = lane_bcast(v, p);
                float sg = fast_sigmoid(fmaf(sig_r[p], vp, -sm_r[p]));
                an = fmaf(we_r[p], sg, an);
                ad = fmaf(w_r[p], sg, ad);
            }
            v = (fmaf(cmt, v, glvl) + an) / (cmt + gl + ad + EPS_);
        }
        if (lane == 0)
            outs[(size_t)b * T_ + t] = fmaf(v, ow, ob);   // MOTOR == 1
    }
}

// ---------------------------------------------------------------------------
// Kernel 3: final FC  (256 x 2048) @ (2048 x 10) + bias  via
// V_WMMA_F32_16X16X4_F32.  One wave per 16-row M tile, 512 K-steps.
// A layout (ISA 7.12.2, 32-bit A 16x4): lane<16 -> K=j, lane>=16 -> K=2+j.
// B (4x16) follows C-matrix striping: VGPR j, lane<16 -> K=j, lane>=16 -> K=2+j.
// C/D: lane%16 = N, VGPR r -> M = r + (lane<16 ? 0 : 8).
// ---------------------------------------------------------------------------
__global__ __launch_bounds__(32)
void fc_wmma(const float* __restrict__ outs,
             const float* __restrict__ fc_w,
             const float* __restrict__ fc_b,
             float* __restrict__ out) {
    const int lane = threadIdx.x;
    const int m0   = blockIdx.x * 16;
    const int rowA = m0 + (lane & 15);
    const int koff = (lane < 16) ? 0 : 2;
    const int nIdx = lane & 15;
    const bool nv  = (nIdx < CLASSES_);
    const int  nc  = nv ? nIdx : 0;                 // clamp to avoid OOB loads
    const float nm = nv ? 1.0f : 0.0f;              // zero out padded columns

    const float* arow = outs + (size_t)rowA * T_;
    const float* brow = fc_w + (size_t)nc * T_;

    v8f c = {};
    for (int k0 = 0; k0 < T_; k0 += 4) {
        v2f a, b;
        a.x = arow[k0 + koff];
        a.y = arow[k0 + koff + 1];
        b.x = brow[k0 + koff] * nm;
        b.y = brow[k0 + koff + 1] * nm;
        c = __builtin_amdgcn_wmma_f32_16x16x4_f32(
                /*neg_a=*/false, a, /*neg_b=*/false, b,
                /*c_mod=*/(short)0, c, /*reuse_a=*/false, /*reuse_b=*/false);
    }

    if (nv) {
        const int mbase = m0 + ((lane < 16) ? 0 : 8);
        const float bias = fc_b[nIdx];
#pragma unroll
        for (int r = 0; r < 8; ++r)
            out[(size_t)(mbase + r) * CLASSES_ + nIdx] = c[r] + bias;
    }
}

// ---------------------------------------------------------------------------
extern "C" void kernel_launch(void* const* d_in, const int* in_sizes, int n_in,
                              void* d_out, int out_size, void* d_ws, size_t ws_size,
                              hipStream_t stream) {
    const float* x        = (const float*)d_in[0];
    const float* input_w  = (const float*)d_in[1];
    const float* input_b  = (const float*)d_in[2];
    const float* smu      = (const float*)d_in[3];
    const float* ssig     = (const float*)d_in[4];
    const float* sw       = (const float*)d_in[5];
    const float* serev    = (const float*)d_in[6];
    const float* mu       = (const float*)d_in[7];
    const float* sigma    = (const float*)d_in[8];
    const float* w        = (const float*)d_in[9];
    const float* erev     = (const float*)d_in[10];
    const float* gleak    = (const float*)d_in[11];
    const float* vleak    = (const float*)d_in[12];
    const float* cm       = (const float*)d_in[13];
    const float* output_w = (const float*)d_in[14];
    const float* output_b = (const float*)d_in[15];
    const float* fc_w     = (const float*)d_in[16];
    const float* fc_b     = (const float*)d_in[17];
    const int*   smask    = (const int*)d_in[18];
    const int*   mask     = (const int*)d_in[19];
    float* out = (float*)d_out;

    float* ws = (float*)d_ws;
    const size_t BT = (size_t)B_ * T_;
    const size_t need_pre = (2 * BT * N_ + BT) * sizeof(float);   // ~82 MB
    const bool pre = (ws_size >= need_pre);

    float *wns = nullptr, *wds = nullptr, *outs;
    if (pre) {
        wns  = ws;
        wds  = ws + BT * N_;
        outs = ws + 2 * BT * N_;
    } else {
        outs = ws;   // fused path needs only B*T floats (2 MB)
    }

    if (pre) {
        const size_t total = BT * N_;
        const int blk = 256;
        const int grd = (int)((total + blk - 1) / blk);
        sensory_precompute<<<grd, blk, 0, stream>>>(x, input_w, input_b, smu,
                                                    ssig, sw, serev, smask,
                                                    wns, wds);
        ltc_scan<true><<<B_, 32, 0, stream>>>(x, input_w, input_b, smu, ssig,
                                              sw, serev, smask, mu, sigma, w,
                                              erev, mask, gleak, vleak, cm,
                                              output_w, output_b, wns, wds,
                                              outs);
    } else {
        ltc_scan<false><<<B_, 32, 0, stream>>>(x, input_w, input_b, smu, ssig,
                                               sw, serev, smask, mu, sigma, w,
                                               erev, mask, gleak, vleak, cm,
                                               output_w, output_b, nullptr,
                                               nullptr, outs);
    }

    fc_wmma<<<B_ / 16, 32, 0, stream>>>(outs, fc_w, fc_b, out);
}